// Net_14370960573240
// MI455X (gfx1250) — compile-verified
//
#include <hip/hip_runtime.h>
#include <math.h>

typedef float v8f __attribute__((ext_vector_type(8)));
typedef float v2f __attribute__((ext_vector_type(2)));

#define F_IN 1346
#define DD   32

// ---------------------------------------------------------------------------
// helpers
// ---------------------------------------------------------------------------
__device__ __forceinline__ float lrelu02(float v) {
    // leaky_relu slope 0.2:  max(v, 0.2v) is exact for both signs
    return fmaxf(v, 0.2f * v);
}

__device__ __forceinline__ void atomicMaxF(float* addr, float val) {
    // classic sign-split trick: int-max for >=0, uint-min for <0
    if (val >= 0.0f) atomicMax((int*)addr, __float_as_int(val));
    else             atomicMin((unsigned int*)addr, __float_as_uint(val));
}

// ---------------------------------------------------------------------------
// K1: h0 = relu(x[N,1346] @ W0[1346,32] + b0)   (fp32 WMMA 16x16x4)
// one wave -> 16 rows x 32 cols (two 16x16 tiles)
// main K-loop is branch-free; the 2-element K remainder is handled by a
// single masked tail step (index clamp + 0/1 multiply, no divergent loads)
// ---------------------------------------------------------------------------
__global__ __launch_bounds__(256)
void wmma_gemm_relu_k1346(const float* __restrict__ A, const float* __restrict__ B,
                          const float* __restrict__ bias, float* __restrict__ C, int nrows) {
    int wave = (blockIdx.x * blockDim.x + threadIdx.x) >> 5;
    int lane = threadIdx.x & 31;
    int row0 = wave * 16;
    if (row0 >= nrows) return;                    // wave-uniform: EXEC stays full
    int half = lane >> 4;                         // 0: K pair 0/1, 1: K pair 2/3
    int m    = lane & 15;

    const float* arow = A + (size_t)(row0 + m) * F_IN;
    const float* aptr = arow + 2 * half;          // this lane's K offset
    v8f acc0 = {}; v8f acc1 = {};

    #pragma unroll 4
    for (int k = 0; k < 1344; k += 4) {
        int ka = k + 2 * half;
        v2f a = *(const v2f*)(aptr + k);          // global_load_b64, 8B aligned
        v2f b0v, b1v;
        b0v.x = B[ka * DD + m];        b0v.y = B[(ka + 1) * DD + m];
        b1v.x = B[ka * DD + 16 + m];   b1v.y = B[(ka + 1) * DD + 16 + m];
        acc0 = __builtin_amdgcn_wmma_f32_16x16x4_f32(false, a, false, b0v, (short)0, acc0, false, false);
        acc1 = __builtin_amdgcn_wmma_f32_16x16x4_f32(false, a, false, b1v, (short)0, acc1, false, false);
    }

    // tail step: k = 1344 (valid K pair only for half==0); branch-free
    {
        float msk = (half == 0) ? 1.0f : 0.0f;    // zero A for out-of-range lanes
        int   ka  = (half == 0) ? 1344 : 0;       // clamped, always in-bounds
        v2f a, b0v, b1v;
        a.x   = arow[ka] * msk;        a.y   = arow[ka + 1] * msk;
        b0v.x = B[ka * DD + m];        b0v.y = B[(ka + 1) * DD + m];
        b1v.x = B[ka * DD + 16 + m];   b1v.y = B[(ka + 1) * DD + 16 + m];
        acc0 = __builtin_amdgcn_wmma_f32_16x16x4_f32(false, a, false, b0v, (short)0, acc0, false, false);
        acc1 = __builtin_amdgcn_wmma_f32_16x16x4_f32(false, a, false, b1v, (short)0, acc1, false, false);
    }

    float bi0 = bias[m], bi1 = bias[16 + m];
    #pragma unroll
    for (int j = 0; j < 8; ++j) {
        int r = row0 + j + 8 * half;              // D layout: VGPR j -> M=j (+8 upper lanes)
        C[(size_t)r * DD + m]      = fmaxf(acc0[j] + bi0, 0.0f);
        C[(size_t)r * DD + 16 + m] = fmaxf(acc1[j] + bi1, 0.0f);
    }
}

// ---------------------------------------------------------------------------
// K2: fused  hg = h @ Wg   and   hres = h @ Wr + br      (K = 32, fp32 WMMA)
// ---------------------------------------------------------------------------
__global__ __launch_bounds__(256)
void wmma_dual32(const float* __restrict__ H, const float* __restrict__ Wg,
                 const float* __restrict__ Wr, const float* __restrict__ br,
                 float* __restrict__ HG, float* __restrict__ HR, int nrows) {
    int wave = (blockIdx.x * blockDim.x + threadIdx.x) >> 5;
    int lane = threadIdx.x & 31;
    int row0 = wave * 16;
    if (row0 >= nrows) return;
    int half = lane >> 4;
    int m    = lane & 15;

    const float* arow = H + (size_t)(row0 + m) * DD;
    v8f ag0 = {}, ag1 = {}, ar0 = {}, ar1 = {};

    #pragma unroll
    for (int k = 0; k < DD; k += 4) {
        int ka = k + 2 * half;
        v2f a = *(const v2f*)(arow + ka);
        v2f g0, g1, r0, r1;
        g0.x = Wg[ka * DD + m];        g0.y = Wg[(ka + 1) * DD + m];
        g1.x = Wg[ka * DD + 16 + m];   g1.y = Wg[(ka + 1) * DD + 16 + m];
        r0.x = Wr[ka * DD + m];        r0.y = Wr[(ka + 1) * DD + m];
        r1.x = Wr[ka * DD + 16 + m];   r1.y = Wr[(ka + 1) * DD + 16 + m];
        ag0 = __builtin_amdgcn_wmma_f32_16x16x4_f32(false, a, false, g0, (short)0, ag0, false, false);
        ag1 = __builtin_amdgcn_wmma_f32_16x16x4_f32(false, a, false, g1, (short)0, ag1, false, false);
        ar0 = __builtin_amdgcn_wmma_f32_16x16x4_f32(false, a, false, r0, (short)0, ar0, false, false);
        ar1 = __builtin_amdgcn_wmma_f32_16x16x4_f32(false, a, false, r1, (short)0, ar1, false, false);
    }

    float bi0 = br[m], bi1 = br[16 + m];
    #pragma unroll
    for (int j = 0; j < 8; ++j) {
        int r = row0 + j + 8 * half;
        HG[(size_t)r * DD + m]      = ag0[j];
        HG[(size_t)r * DD + 16 + m] = ag1[j];
        HR[(size_t)r * DD + m]      = ar0[j] + bi0;
        HR[(size_t)r * DD + 16 + m] = ar1[j] + bi1;
    }
}

// ---------------------------------------------------------------------------
// K3: per (node,head): attention logits + init m with self-loop logit
// ---------------------------------------------------------------------------
__global__ __launch_bounds__(256)
void logits_kernel(const float* __restrict__ hg, const float* __restrict__ asrc,
                   const float* __restrict__ adst, float* __restrict__ al_s,
                   float* __restrict__ al_d, float* __restrict__ m, int n) {
    int g = blockIdx.x * blockDim.x + threadIdx.x;
    int i = g >> 2, h = g & 3;
    if (i >= n) return;
    float ss = 0.0f, dd = 0.0f;
    #pragma unroll
    for (int c = 0; c < 8; ++c) {
        float hv = hg[(size_t)i * DD + h * 8 + c];
        ss += hv * asrc[h * 8 + c];
        dd += hv * adst[h * 8 + c];
    }
    al_s[i * 4 + h] = ss;
    al_d[i * 4 + h] = dd;
    m[i * 4 + h]    = lrelu02(ss + dd);   // self-loop seeds the segment max
}

// ---------------------------------------------------------------------------
// K4: per (edge,head): segment max via atomics
// ---------------------------------------------------------------------------
__global__ __launch_bounds__(256)
void edge_max_kernel(const int* __restrict__ src, const int* __restrict__ dst,
                     const float* __restrict__ al_s, const float* __restrict__ al_d,
                     float* __restrict__ m, int E) {
    int g = blockIdx.x * blockDim.x + threadIdx.x;
    int e = g >> 2, h = g & 3;
    if (e >= E) return;
    int s = src[e], d = dst[e];
    float v = lrelu02(al_s[s * 4 + h] + al_d[d * 4 + h]);
    atomicMaxF(&m[d * 4 + h], v);
}

// ---------------------------------------------------------------------------
// K5: per (node,channel): seed denom/acc with the self-loop term
// ---------------------------------------------------------------------------
__global__ __launch_bounds__(256)
void selfloop_kernel(const float* __restrict__ hg, const float* __restrict__ al_s,
                     const float* __restrict__ al_d, const float* __restrict__ m,
                     float* __restrict__ denom, float* __restrict__ acc, int n) {
    int g = blockIdx.x * blockDim.x + threadIdx.x;
    int i = g >> 5, ch = g & 31, h = ch >> 3;
    if (i >= n) return;
    float v  = lrelu02(al_s[i * 4 + h] + al_d[i * 4 + h]);
    float ex = expf(v - m[i * 4 + h]);
    if ((ch & 7) == 0) denom[i * 4 + h] = ex;
    acc[(size_t)i * DD + ch] = ex * hg[(size_t)i * DD + ch];
}

// ---------------------------------------------------------------------------
// K6: per (edge,channel): accumulate unnormalized softmax-weighted messages
// ---------------------------------------------------------------------------
__global__ __launch_bounds__(256)
void edge_acc_kernel(const int* __restrict__ src, const int* __restrict__ dst,
                     const float* __restrict__ al_s, const float* __restrict__ al_d,
                     const float* __restrict__ m, const float* __restrict__ hg,
                     float* __restrict__ denom, float* __restrict__ acc, int E) {
    int g = blockIdx.x * blockDim.x + threadIdx.x;
    int e = g >> 5, ch = g & 31, h = ch >> 3;
    if (e >= E) return;
    int s = src[e], d = dst[e];
    float v  = lrelu02(al_s[s * 4 + h] + al_d[d * 4 + h]);
    float ex = expf(v - m[d * 4 + h]);
    if ((ch & 7) == 0) atomicAdd(&denom[d * 4 + h], ex);
    atomicAdd(&acc[(size_t)d * DD + ch], ex * hg[(size_t)s * DD + ch]);
}

// ---------------------------------------------------------------------------
// K7: per (node,channel): normalize, + bias + residual, ELU
// ---------------------------------------------------------------------------
__global__ __launch_bounds__(256)
void finalize_kernel(const float* __restrict__ acc, const float* __restrict__ denom,
                     const float* __restrict__ bg, const float* __restrict__ hres,
                     float* __restrict__ hout, int n) {
    int g = blockIdx.x * blockDim.x + threadIdx.x;
    int i = g >> 5, ch = g & 31, h = ch >> 3;
    if (i >= n) return;
    float v = acc[(size_t)i * DD + ch] / (denom[i * 4 + h] + 1e-16f)
            + bg[ch] + hres[(size_t)i * DD + ch];
    hout[(size_t)i * DD + ch] = (v > 0.0f) ? v : (expf(v) - 1.0f);
}

// ---------------------------------------------------------------------------
// K8: per node: z = relu([h0|h1|h2] @ Wfc + bfc); out = sigmoid(z @ Wout + bout)
// ---------------------------------------------------------------------------
__global__ __launch_bounds__(256)
void head_kernel(const float* __restrict__ h0, const float* __restrict__ h1,
                 const float* __restrict__ h2, const float* __restrict__ Wfc,
                 const float* __restrict__ bfc, const float* __restrict__ Wout,
                 const float* __restrict__ bout, float* __restrict__ out, int n) {
    int i = blockIdx.x * blockDim.x + threadIdx.x;
    if (i >= n) return;
    float z[16];
    #pragma unroll
    for (int j = 0; j < 16; ++j) z[j] = bfc[j];
    for (int k = 0; k < 96; ++k) {
        float hv = (k < 32) ? h0[(size_t)i * DD + k]
                 : (k < 64) ? h1[(size_t)i * DD + (k - 32)]
                            : h2[(size_t)i * DD + (k - 64)];
        #pragma unroll
        for (int j = 0; j < 16; ++j) z[j] += hv * Wfc[k * 16 + j];
    }
    float o = bout[0];
    #pragma unroll
    for (int j = 0; j < 16; ++j) o += fmaxf(z[j], 0.0f) * Wout[j];
    out[i] = 1.0f / (1.0f + expf(-o));
}

// ---------------------------------------------------------------------------
// launcher
// ---------------------------------------------------------------------------
extern "C" void kernel_launch(void* const* d_in, const int* in_sizes, int n_in,
                              void* d_out, int out_size, void* d_ws, size_t ws_size,
                              hipStream_t stream) {
    const float* x     = (const float*)d_in[0];
    const int*   ei    = (const int*)  d_in[1];
    const float* W0    = (const float*)d_in[2];
    const float* b0    = (const float*)d_in[3];
    const float* Wg1   = (const float*)d_in[4];
    const float* asrc1 = (const float*)d_in[5];
    const float* adst1 = (const float*)d_in[6];
    const float* bg1   = (const float*)d_in[7];
    const float* Wr1   = (const float*)d_in[8];
    const float* br1   = (const float*)d_in[9];
    const float* Wg2   = (const float*)d_in[10];
    const float* asrc2 = (const float*)d_in[11];
    const float* adst2 = (const float*)d_in[12];
    const float* bg2   = (const float*)d_in[13];
    const float* Wr2   = (const float*)d_in[14];
    const float* br2   = (const float*)d_in[15];
    const float* Wfc   = (const float*)d_in[16];
    const float* bfc   = (const float*)d_in[17];
    const float* Wout  = (const float*)d_in[18];
    const float* bout  = (const float*)d_in[19];

    const int N = in_sizes[0] / F_IN;
    const int E = in_sizes[1] / 2;
    const int* src = ei;
    const int* dst = ei + E;

    float* ws = (float*)d_ws;
    float* h0    = ws;  ws += (size_t)N * DD;
    float* h1    = ws;  ws += (size_t)N * DD;
    float* h2    = ws;  ws += (size_t)N * DD;
    float* hg    = ws;  ws += (size_t)N * DD;
    float* hres  = ws;  ws += (size_t)N * DD;
    float* acc   = ws;  ws += (size_t)N * DD;
    float* al_s  = ws;  ws += (size_t)N * 4;
    float* al_d  = ws;  ws += (size_t)N * 4;
    float* mbuf  = ws;  ws += (size_t)N * 4;
    float* denom = ws;  ws += (size_t)N * 4;

    const int BS = 256;
    int tiles      = (N + 15) / 16;
    int gemmBlocks = (tiles * 32 + BS - 1) / BS;
    int n4Blocks   = (N * 4  + BS - 1) / BS;
    int n32Blocks  = (N * 32 + BS - 1) / BS;
    int e4Blocks   = (int)(((size_t)E * 4  + BS - 1) / BS);
    int e32Blocks  = (int)(((size_t)E * 32 + BS - 1) / BS);
    int nBlocks    = (N + BS - 1) / BS;

    // h0 = relu(x @ W0 + b0)
    wmma_gemm_relu_k1346<<<gemmBlocks, BS, 0, stream>>>(x, W0, b0, h0, N);

    const float* layerIn[2]  = {h0, h1};
    float*       layerOut[2] = {h1, h2};
    const float* Wg[2]   = {Wg1, Wg2};
    const float* Wr[2]   = {Wr1, Wr2};
    const float* br[2]   = {br1, br2};
    const float* bg[2]   = {bg1, bg2};
    const float* as_[2]  = {asrc1, asrc2};
    const float* ad_[2]  = {adst1, adst2};

    for (int L = 0; L < 2; ++L) {
        wmma_dual32<<<gemmBlocks, BS, 0, stream>>>(layerIn[L], Wg[L], Wr[L], br[L], hg, hres, N);
        logits_kernel<<<n4Blocks, BS, 0, stream>>>(hg, as_[L], ad_[L], al_s, al_d, mbuf, N);
        edge_max_kernel<<<e4Blocks, BS, 0, stream>>>(src, dst, al_s, al_d, mbuf, E);
        selfloop_kernel<<<n32Blocks, BS, 0, stream>>>(hg, al_s, al_d, mbuf, denom, acc, N);
        edge_acc_kernel<<<e32Blocks, BS, 0, stream>>>(src, dst, al_s, al_d, mbuf, hg, denom, acc, E);
        finalize_kernel<<<n32Blocks, BS, 0, stream>>>(acc, denom, bg[L], hres, layerOut[L], N);
    }

    head_kernel<<<nBlocks, BS, 0, stream>>>(h0, h1, h2, Wfc, bfc, Wout, bout, (float*)d_out, N);
}